// GCNNet_65111704207441
// MI455X (gfx1250) — compile-verified
//
#include <hip/hip_runtime.h>
#include <hip/hip_bf16.h>
#include <math.h>

// ---------------------------------------------------------------------------
// GCN 3-layer forward for gfx1250 (MI455X).
//   * GEMMs: V_WMMA_F32_16X16X4_F32, one wave = 16x128 row-strip (8 acc tiles),
//     compile-time K/DOUT -> fully immediate-offset loads, dense WMMA stream.
//   * Edge aggregation: warp-per-edge global_atomic_add_f32; h/agg arrays are
//     L2-resident (25.6 MB << 192 MB) so the scatter runs at L2 bandwidth.
// ---------------------------------------------------------------------------

typedef __attribute__((ext_vector_type(2))) float v2f;
typedef __attribute__((ext_vector_type(8))) float v8f;

#define D_IN  128
#define D_H   128
#define D_OUT 10

// ---------------- degree / normalization ----------------
__global__ void k_init_deg(float* __restrict__ deg, int n) {
    int i = blockIdx.x * blockDim.x + threadIdx.x;
    if (i < n) deg[i] = 1.0f;   // self-loop
}

__global__ void k_count_deg(const long long* __restrict__ dst,
                            float* __restrict__ deg, int E) {
    int i = blockIdx.x * blockDim.x + threadIdx.x;
    if (i < E) atomicAdd(&deg[(int)dst[i]], 1.0f);
}

__global__ void k_rsqrt_inplace(float* __restrict__ deg, int n) {
    int i = blockIdx.x * blockDim.x + threadIdx.x;
    if (i < n) deg[i] = rsqrtf(deg[i]);   // dis = 1/sqrt(deg)
}

// ---------------- fp32 WMMA GEMM: H[nRows x DOUT] = A[nRows x K] @ W[K x DOUT]
// One wave computes a 16-row x DOUT strip: CT = ceil(DOUT/16) accumulator tiles.
// Per K-step (4): one A load feeds CT consecutive WMMAs (register blocking).
// Layouts per CDNA5 ISA 7.12.2 (32-bit A 16x4, B 4x16, C/D 16x16).
template<int K, int DOUT>
__global__ __launch_bounds__(256)
void k_gemm_wmma_strip(const float* __restrict__ A, const float* __restrict__ W,
                       float* __restrict__ H, int rowTiles) {
    constexpr int CT = (DOUT + 15) / 16;

    const int wave = (int)((blockIdx.x * blockDim.x + threadIdx.x) >> 5);
    const int lane = threadIdx.x & 31;
    if (wave >= rowTiles) return;             // wave-uniform: EXEC all-1s inside

    const int row0 = wave << 4;
    const int m    = lane & 15;               // M (A rows) / N (B,C,D cols)
    const int kHi  = (lane >> 4) << 1;        // lanes 16-31 hold K = {2,3}

    // per-lane base pointers; all loop accesses are immediate offsets
    const float* ap = A + (size_t)(row0 + m) * K + kHi;

    // B column index per tile; clamp into bounds when DOUT % 16 != 0
    // (garbage columns never influence valid outputs and are never stored)
    const float* wp[CT];
#pragma unroll
    for (int t = 0; t < CT; ++t) {
        int ncol  = t * 16 + m;
        int ncolc = (DOUT % 16 == 0) ? ncol : (ncol < DOUT ? ncol : DOUT - 1);
        wp[t] = W + (size_t)kHi * DOUT + ncolc;
    }

    v8f acc[CT];
#pragma unroll
    for (int t = 0; t < CT; ++t) acc[t] = (v8f){};

#pragma unroll 4
    for (int k0 = 0; k0 < K; k0 += 4) {
        v2f a;                                 // A: contiguous pair -> b64 load
        a.x = ap[k0];
        a.y = ap[k0 + 1];
#pragma unroll
        for (int t = 0; t < CT; ++t) {
            v2f b;
            b.x = wp[t][(size_t)k0 * DOUT];
            b.y = wp[t][(size_t)(k0 + 1) * DOUT];
            acc[t] = __builtin_amdgcn_wmma_f32_16x16x4_f32(
                        /*neg_a=*/false, a, /*neg_b=*/false, b,
                        /*c_mod=*/(short)0, acc[t],
                        /*reuse_a=*/false, /*reuse_b=*/false);
        }
    }

    // C/D: VGPR r -> M = r (lanes 0-15) or r+8 (lanes 16-31), N = lane&15
    const int rbase = row0 + ((lane >> 4) << 3);
#pragma unroll
    for (int t = 0; t < CT; ++t) {
        int ncol = t * 16 + m;
        if (DOUT % 16 == 0 || ncol < DOUT) {
#pragma unroll
            for (int r = 0; r < 8; ++r)
                H[(size_t)(rbase + r) * DOUT + ncol] = acc[t][r];
        }
    }
}

// ---------------- agg = h * dis^2 + bias (folds self-loop + bias, no memset)
template<int D>
__global__ void k_init_agg(const float* __restrict__ h, const float* __restrict__ dis,
                           const float* __restrict__ bias, float* __restrict__ agg,
                           int n) {
    int i = blockIdx.x * blockDim.x + threadIdx.x;
    if (i >= n * D) return;
    int node = i / D;           // D compile-time: shift (128) / magic-mul (10)
    int j    = i - node * D;
    float d  = dis[node];
    agg[i] = h[i] * d * d + bias[j];
}

// ---------------- edge scatter, D = 128: warp per edge, float4 per lane
__global__ __launch_bounds__(256)
void k_scatter128(const float* __restrict__ h, const float* __restrict__ dis,
                  const long long* __restrict__ src, const long long* __restrict__ dst,
                  float* __restrict__ agg, int E) {
    int e    = (int)((blockIdx.x * blockDim.x + threadIdx.x) >> 5);
    int lane = threadIdx.x & 31;
    if (e >= E) return;
    int s = (int)src[e];
    int d = (int)dst[e];
    float norm = dis[s] * dis[d];
    const float4* hp = (const float4*)(h + (size_t)s * 128);   // b128 gather
    float*        ap = agg + (size_t)d * 128 + lane * 4;
    float4 v = hp[lane];
    atomicAdd(ap + 0, v.x * norm);
    atomicAdd(ap + 1, v.y * norm);
    atomicAdd(ap + 2, v.z * norm);
    atomicAdd(ap + 3, v.w * norm);
}

// ---------------- edge scatter, generic D (layer 3: D = 10)
__global__ __launch_bounds__(256)
void k_scatter(const float* __restrict__ h, const float* __restrict__ dis,
               const long long* __restrict__ src, const long long* __restrict__ dst,
               float* __restrict__ agg, int E, int D) {
    int e    = (int)((blockIdx.x * blockDim.x + threadIdx.x) >> 5);
    int lane = threadIdx.x & 31;
    if (e >= E) return;
    int s = (int)src[e];
    int d = (int)dst[e];
    float norm = dis[s] * dis[d];
    const float* hp = h   + (size_t)s * D;
    float*       ap = agg + (size_t)d * D;
    for (int j = lane; j < D; j += 32)
        atomicAdd(&ap[j], hp[j] * norm);
}

__global__ void k_relu_inplace(float* __restrict__ x, int n) {
    int i = blockIdx.x * blockDim.x + threadIdx.x;
    if (i < n) x[i] = fmaxf(x[i], 0.0f);
}

// ---------------- final: relu then log_softmax over 10 classes, per node
__global__ void k_relu_logsoftmax(const float* __restrict__ agg,
                                  float* __restrict__ out, int n) {
    int i = blockIdx.x * blockDim.x + threadIdx.x;
    if (i >= n) return;
    float v[D_OUT];
    float mx = -INFINITY;
#pragma unroll
    for (int j = 0; j < D_OUT; ++j) {
        float t = fmaxf(agg[(size_t)i * D_OUT + j], 0.0f);
        v[j] = t;
        mx = fmaxf(mx, t);
    }
    float sum = 0.0f;
#pragma unroll
    for (int j = 0; j < D_OUT; ++j) sum += expf(v[j] - mx);
    float lse = mx + logf(sum);
#pragma unroll
    for (int j = 0; j < D_OUT; ++j)
        out[(size_t)i * D_OUT + j] = v[j] - lse;
}

// ---------------------------------------------------------------------------
static inline int cdiv(int a, int b) { return (a + b - 1) / b; }

extern "C" void kernel_launch(void* const* d_in, const int* in_sizes, int n_in,
                              void* d_out, int out_size, void* d_ws, size_t ws_size,
                              hipStream_t stream) {
    const float*     x    = (const float*)d_in[0];
    const long long* eidx = (const long long*)d_in[1];   // int64, shape [2, E]
    const float*     W1   = (const float*)d_in[2];
    const float*     b1   = (const float*)d_in[3];
    const float*     W2   = (const float*)d_in[4];
    const float*     b2   = (const float*)d_in[5];
    const float*     W3   = (const float*)d_in[6];
    const float*     b3   = (const float*)d_in[7];
    float*           out  = (float*)d_out;

    const int N = in_sizes[0] / D_IN;        // 50000
    const int E = in_sizes[1] / 2;           // 800000
    const long long* e_src = eidx;
    const long long* e_dst = eidx + E;

    // workspace layout (floats): dis[N pad] | h[N*128] | agg[N*128]  (~51.5 MB)
    float* ws = (float*)d_ws;
    const size_t disPad = ((size_t)N + 127) & ~(size_t)127;
    float* dis = ws;
    float* h   = ws + disPad;
    float* agg = h + (size_t)N * D_H;

    const int TB = 256;

    // ---- normalization (computed once, reused by all 3 layers) ----
    k_init_deg     <<<cdiv(N, TB), TB, 0, stream>>>(dis, N);
    k_count_deg    <<<cdiv(E, TB), TB, 0, stream>>>(e_dst, dis, E);
    k_rsqrt_inplace<<<cdiv(N, TB), TB, 0, stream>>>(dis, N);

    const int rowTiles   = N / 16;                        // 3125, exact
    const int gemmBlocks = cdiv(rowTiles * 32, TB);
    const int edgeBlocks = cdiv(E * 32, TB);

    // ---- layer 1: x[N,128] @ W1[128,128] ----
    k_gemm_wmma_strip<D_IN, D_H><<<gemmBlocks, TB, 0, stream>>>(x, W1, h, rowTiles);
    k_init_agg<D_H>  <<<cdiv(N * D_H, TB), TB, 0, stream>>>(h, dis, b1, agg, N);
    k_scatter128     <<<edgeBlocks, TB, 0, stream>>>(h, dis, e_src, e_dst, agg, E);
    k_relu_inplace   <<<cdiv(N * D_H, TB), TB, 0, stream>>>(agg, N * D_H);

    // ---- layer 2: agg[N,128] @ W2[128,128] (gemm consumes agg, then agg reused)
    k_gemm_wmma_strip<D_H, D_H><<<gemmBlocks, TB, 0, stream>>>(agg, W2, h, rowTiles);
    k_init_agg<D_H>  <<<cdiv(N * D_H, TB), TB, 0, stream>>>(h, dis, b2, agg, N);
    k_scatter128     <<<edgeBlocks, TB, 0, stream>>>(h, dis, e_src, e_dst, agg, E);
    k_relu_inplace   <<<cdiv(N * D_H, TB), TB, 0, stream>>>(agg, N * D_H);

    // ---- layer 3: agg[N,128] @ W3[128,10] -> h (reused, N*10 fits) ----
    k_gemm_wmma_strip<D_H, D_OUT><<<gemmBlocks, TB, 0, stream>>>(agg, W3, h, rowTiles);
    k_init_agg<D_OUT><<<cdiv(N * D_OUT, TB), TB, 0, stream>>>(h, dis, b3, agg, N);
    k_scatter        <<<edgeBlocks, TB, 0, stream>>>(h, dis, e_src, e_dst, agg, E, D_OUT);
    k_relu_logsoftmax<<<cdiv(N, TB), TB, 0, stream>>>(agg, out, N);

    (void)n_in; (void)out_size; (void)ws_size;
}